// SlotAttentionTransformer_14259291423096
// MI455X (gfx1250) — compile-verified
//
#include <hip/hip_runtime.h>
#include <math.h>

typedef float v2f __attribute__((ext_vector_type(2)));
typedef float v8f __attribute__((ext_vector_type(8)));

#define NPIX 4096
#define NB   8

__device__ __forceinline__ float block_reduce_256(float v, float* red, int tid) {
  red[tid] = v;
  __syncthreads();
  for (int off = 128; off > 0; off >>= 1) {
    if (tid < off) red[tid] += red[tid + off];
    __syncthreads();
  }
  float r = red[0];
  __syncthreads();
  return r;
}

// ---------------------------------------------------------------------------
// Kernel P: layernorm(feat) fused into 4 DIM x DIM GEMMs via f32 WMMA.
// One wave = 16 rows x 64 out-cols strip; V_WMMA_F32_16X16X4_F32, K-loop of 16.
// B operand stored as K-pairs (float2) so both fragments are ds_load_b64.
// ---------------------------------------------------------------------------
__global__ __launch_bounds__(256) void k_pre(
    const float* __restrict__ feat,
    const float* __restrict__ g_feat, const float* __restrict__ b_feat,
    const float* __restrict__ Wkf, const float* __restrict__ Wvf,
    const float* __restrict__ Wkb, const float* __restrict__ Wvb,
    float* __restrict__ kf, float* __restrict__ vf,
    float* __restrict__ kb, float* __restrict__ vb)
{
  __shared__ v2f   sB[32][64];      // sB[p][d] = (W[d][2p], W[d][2p+1])  16KB
  __shared__ float sA[8][16][64];   // per-wave layer-normed tiles        32KB
  const int tid  = threadIdx.x;
  const int wave = tid >> 5;
  const int lane = tid & 31;
  const int rowbase = blockIdx.x * 128 + wave * 16;   // flat row over (b,n)

  // stage 16x64 activation tile with b128 loads/stores
  {
    const float4* src = (const float4*)(feat + (size_t)rowbase * 64);
    float4* dst = (float4*)(&sA[wave][0][0]);
    for (int i = 0; i < 8; ++i) dst[i * 32 + lane] = src[i * 32 + lane];
  }
  __syncthreads();
  // in-place layernorm: lane pair (l, l+16) both own row l&15, write halves
  {
    int r = lane & 15;
    float m = 0.f;
    for (int c = 0; c < 64; ++c) m += sA[wave][r][c];
    m *= 0.015625f;
    float v = 0.f;
    for (int c = 0; c < 64; ++c) { float t = sA[wave][r][c] - m; v += t * t; }
    v *= 0.015625f;
    float inv = 1.0f / sqrtf(v + 1e-5f);
    int c0 = (lane < 16) ? 0 : 32;
    for (int c = c0; c < c0 + 32; ++c)
      sA[wave][r][c] = (sA[wave][r][c] - m) * inv * g_feat[c] + b_feat[c];
  }
  __syncthreads();

  const float* Ws[4] = {Wkf, Wvf, Wkb, Wvb};
  float* Os[4]; Os[0] = kf; Os[1] = vf; Os[2] = kb; Os[3] = vb;
  const int half = lane >> 4;      // 0: K pair (0,1) / D rows 0..7
  const int l    = lane & 15;
  for (int m = 0; m < 4; ++m) {
    const float* W = Ws[m];
    // transpose + pair-pack: W[d][c] -> sB[c>>1][d][c&1]
    for (int i = tid; i < 4096; i += 256) {
      int d = i >> 6, c = i & 63;
      ((float*)&sB[c >> 1][d])[c & 1] = W[i];
    }
    __syncthreads();
    for (int nt = 0; nt < 4; ++nt) {
      v8f acc = {};
      for (int kk = 0; kk < 16; ++kk) {
        int k0 = kk * 4 + half * 2;
        v2f a;
        a.x = sA[wave][l][k0];
        a.y = sA[wave][l][k0 + 1];
        v2f b = sB[kk * 2 + half][nt * 16 + l];   // single ds_load_b64
        acc = __builtin_amdgcn_wmma_f32_16x16x4_f32(
            false, a, false, b, (short)0, acc, false, false);
      }
      float* outp = Os[m];
      for (int r = 0; r < 8; ++r)   // D layout: lane<16 -> M=r, else M=8+r
        outp[(size_t)(rowbase + half * 8 + r) * 64 + nt * 16 + l] = acc[r];
    }
    __syncthreads();
  }
}

// ---------------------------------------------------------------------------
// Kernel A: per (b,slot): LN(slot) @ Wq^T, qt = Wm^T q, alpha/beta/gamma.
// Also initializes slot/position state from the init inputs at iter 0.
// ---------------------------------------------------------------------------
__global__ __launch_bounds__(64) void k_qt(
    const float* __restrict__ Wq_fg, const float* __restrict__ Wq_bg,
    const float* __restrict__ Wm_fg, const float* __restrict__ Wm_bg,
    const float* __restrict__ gq_fg, const float* __restrict__ bq_fg,
    const float* __restrict__ gq_bg, const float* __restrict__ bq_bg,
    const float* __restrict__ W_grid, const float* __restrict__ b_grid,
    const float* __restrict__ sinit_fg, const float* __restrict__ sinit_bg,
    const float* __restrict__ pinit,
    float* __restrict__ slots, float* __restrict__ pos,
    float* __restrict__ qt, float* __restrict__ abc, int iter)
{
  const int b = blockIdx.x >> 3;
  const int s = blockIdx.x & 7;
  const int d = threadIdx.x;
  __shared__ float lnv[64], qv[64], qtv[64], mv[2];
  float* slot = slots + (size_t)(b * 8 + s) * 64;
  if (iter == 0) {
    slot[d] = (s == 0) ? sinit_bg[d] : sinit_fg[(s - 1) * 64 + d];
    if (s > 0 && d < 2) pos[(b * 7 + (s - 1)) * 2 + d] = pinit[(s - 1) * 2 + d];
  }
  __syncthreads();
  float x = slot[d];
  lnv[d] = x;
  __syncthreads();
  if (d == 0) {
    float m = 0.f; for (int c = 0; c < 64; ++c) m += lnv[c];
    m *= 0.015625f;
    float v = 0.f; for (int c = 0; c < 64; ++c) { float t = lnv[c] - m; v += t * t; }
    mv[0] = m; mv[1] = v * 0.015625f;
  }
  __syncthreads();
  const float* g  = s ? gq_fg : gq_bg;
  const float* bb = s ? bq_fg : bq_bg;
  lnv[d] = (x - mv[0]) / sqrtf(mv[1] + 1e-5f) * g[d] + bb[d];
  __syncthreads();
  const float* Wq = s ? Wq_fg : Wq_bg;
  float q = 0.f;
  for (int c = 0; c < 64; ++c) q += lnv[c] * Wq[d * 64 + c];
  qv[d] = q;
  __syncthreads();
  const float* Wm = s ? Wm_fg : Wm_bg;
  float qtd = 0.f;
  for (int c = 0; c < 64; ++c) qtd += qv[c] * Wm[c * 64 + d];
  qtv[d] = qtd;
  qt[(size_t)(b * 8 + s) * 64 + d] = qtd;
  __syncthreads();
  if (d == 0) {
    float A = 0.f, Bv = 0.f, G = 0.f;
    for (int c = 0; c < 64; ++c) {
      float u = W_grid[c * 4 + 0] - W_grid[c * 4 + 2];
      float w = W_grid[c * 4 + 1] - W_grid[c * 4 + 3];
      A += qtv[c] * u; Bv += qtv[c] * w; G += qtv[c] * b_grid[c];
    }
    abc[(b * 8 + s) * 3 + 0] = A;
    abc[(b * 8 + s) * 3 + 1] = Bv;
    abc[(b * 8 + s) * 3 + 2] = G;
  }
}

// ---------------------------------------------------------------------------
// Kernel B: logits -> softmax over 8 slots -> attn (+EPS); per-chunk partial
// sums of attn, attn*gx, attn*gy (two-phase, no float atomics).
// ---------------------------------------------------------------------------
__global__ __launch_bounds__(256) void k_attn(
    const float* __restrict__ kf, const float* __restrict__ kb,
    const float* __restrict__ qt, const float* __restrict__ abc,
    const float* __restrict__ pos,
    float* __restrict__ attn,
    float* __restrict__ Sp, float* __restrict__ Gxp, float* __restrict__ Gyp)
{
  const int b = blockIdx.x >> 4;
  const int chunk = blockIdx.x & 15;
  const int tid = threadIdx.x;
  __shared__ float sqt[8][64];
  __shared__ float sab[8][3];
  __shared__ float spos[7][2];
  __shared__ float red[256];
  for (int i = tid; i < 512; i += 256) sqt[i >> 6][i & 63] = qt[(size_t)b * 512 + i];
  if (tid < 24) sab[tid / 3][tid % 3] = abc[b * 24 + tid];
  if (tid < 14) spos[tid >> 1][tid & 1] = pos[b * 14 + tid];
  __syncthreads();

  const int n = chunk * 256 + tid;
  const float gx = -1.f + (float)(2 * (n & 63) + 1) * 0.015625f;
  const float gy = -1.f + (float)(2 * (n >> 6) + 1) * 0.015625f;
  const float* kfr = kf + ((size_t)b * NPIX + n) * 64;
  const float* kbr = kb + ((size_t)b * NPIX + n) * 64;
  float logit[8];
  for (int s = 0; s < 8; ++s) {
    const float* kr = s ? kfr : kbr;
    float dot = 0.f;
    for (int c = 0; c < 64; ++c) dot += sqt[s][c] * kr[c];
    float rx = gx, ry = gy;
    if (s > 0) { rx -= spos[s - 1][0]; ry -= spos[s - 1][1]; }
    logit[s] = 0.125f * (dot + rx * sab[s][0] + ry * sab[s][1] + sab[s][2]);
  }
  float mx = logit[0];
  for (int s = 1; s < 8; ++s) mx = fmaxf(mx, logit[s]);
  float e[8], sum = 0.f;
  for (int s = 0; s < 8; ++s) { e[s] = expf(logit[s] - mx); sum += e[s]; }
  float inv = 1.f / sum;
  float av[8];
  for (int s = 0; s < 8; ++s) {
    av[s] = e[s] * inv + 1e-8f;
    attn[(size_t)(b * 8 + s) * NPIX + n] = av[s];
  }
  for (int s = 0; s < 8; ++s) {
    float vs = block_reduce_256(av[s], red, tid);
    float vx = block_reduce_256(av[s] * gx, red, tid);
    float vy = block_reduce_256(av[s] * gy, red, tid);
    if (tid == 0) {
      Sp [(b * 16 + chunk) * 8 + s] = vs;
      Gxp[(b * 16 + chunk) * 8 + s] = vx;
      Gyp[(b * 16 + chunk) * 8 + s] = vy;
    }
  }
}

__global__ __launch_bounds__(64) void k_attn_red(
    const float* __restrict__ Sp, const float* __restrict__ Gxp,
    const float* __restrict__ Gyp,
    float* __restrict__ S, float* __restrict__ Gx, float* __restrict__ Gy)
{
  int t = threadIdx.x;            // t = b*8 + s
  int b = t >> 3, s = t & 7;
  float a = 0.f, x = 0.f, y = 0.f;
  for (int c = 0; c < 16; ++c) {
    a += Sp [(b * 16 + c) * 8 + s];
    x += Gxp[(b * 16 + c) * 8 + s];
    y += Gyp[(b * 16 + c) * 8 + s];
  }
  S[t] = a; Gx[t] = x; Gy[t] = y;
}

// ---------------------------------------------------------------------------
// Kernel C: per (b,slot): weighted v-sum, fused ge contribution via (u,w),
// slot += Wm @ x, position momentum, residual gelu MLP (iters 0..2 only).
// ---------------------------------------------------------------------------
__global__ __launch_bounds__(256) void k_update(
    const float* __restrict__ vf, const float* __restrict__ vb,
    const float* __restrict__ attn, const float* __restrict__ S,
    const float* __restrict__ Gx, const float* __restrict__ Gy,
    const float* __restrict__ W_grid, const float* __restrict__ b_grid,
    const float* __restrict__ Wm_fg, const float* __restrict__ Wm_bg,
    const float* __restrict__ gm_fg, const float* __restrict__ bm_fg,
    const float* __restrict__ gm_bg, const float* __restrict__ bm_bg,
    const float* __restrict__ W1f, const float* __restrict__ b1f,
    const float* __restrict__ W2f, const float* __restrict__ b2f,
    const float* __restrict__ W1b, const float* __restrict__ b1b,
    const float* __restrict__ W2b, const float* __restrict__ b2b,
    float* __restrict__ slots, float* __restrict__ pos)
{
  const int b = blockIdx.x >> 3;
  const int s = blockIdx.x & 7;
  const int tid = threadIdx.x;
  const int g = tid >> 6, d = tid & 63;
  __shared__ float part[4][64];
  __shared__ float xv[64], snew[64], hv[64], h1[64], mv[2];
  const float* vsrc = s ? vf : vb;
  const float* at = attn + (size_t)(b * 8 + s) * NPIX;
  const float invS = 1.0f / S[b * 8 + s];
  float acc = 0.f;
  for (int n = g; n < NPIX; n += 4) {
    __builtin_prefetch(&vsrc[((size_t)b * NPIX + n + 16) * 64 + d], 0, 0);
    acc += at[n] * vsrc[((size_t)b * NPIX + n) * 64 + d];
  }
  part[g][d] = acc;
  __syncthreads();
  const float mxp = Gx[b * 8 + s] * invS;
  const float myp = Gy[b * 8 + s] * invS;
  if (g == 0) {
    float A = (part[0][d] + part[1][d] + part[2][d] + part[3][d]) * invS;
    float rx = mxp, ry = myp;
    if (s > 0) { rx -= pos[(b * 7 + s - 1) * 2 + 0]; ry -= pos[(b * 7 + s - 1) * 2 + 1]; }
    float u = W_grid[d * 4 + 0] - W_grid[d * 4 + 2];
    float w = W_grid[d * 4 + 1] - W_grid[d * 4 + 3];
    xv[d] = A + rx * u + ry * w + b_grid[d];
  }
  __syncthreads();
  if (tid == 0 && s > 0) {   // momentum update (after xv consumed old pos)
    pos[(b * 7 + s - 1) * 2 + 0] = 0.5f * mxp + 0.5f * pos[(b * 7 + s - 1) * 2 + 0];
    pos[(b * 7 + s - 1) * 2 + 1] = 0.5f * myp + 0.5f * pos[(b * 7 + s - 1) * 2 + 1];
  }
  const float* Wm = s ? Wm_fg : Wm_bg;
  if (g == 0) {
    float upd = 0.f;
    for (int c = 0; c < 64; ++c) upd += Wm[d * 64 + c] * xv[c];
    snew[d] = slots[(size_t)(b * 8 + s) * 64 + d] + upd;
  }
  __syncthreads();
  if (tid == 0) {
    float m = 0.f; for (int c = 0; c < 64; ++c) m += snew[c];
    m *= 0.015625f;
    float v = 0.f; for (int c = 0; c < 64; ++c) { float t = snew[c] - m; v += t * t; }
    mv[0] = m; mv[1] = v * 0.015625f;
  }
  __syncthreads();
  const float* gm = s ? gm_fg : gm_bg;
  const float* bm = s ? bm_fg : bm_bg;
  if (g == 0)
    hv[d] = (snew[d] - mv[0]) / sqrtf(mv[1] + 1e-5f) * gm[d] + bm[d];
  __syncthreads();
  const float* W1 = s ? W1f : W1b;  const float* B1 = s ? b1f : b1b;
  const float* W2 = s ? W2f : W2b;  const float* B2 = s ? b2f : b2b;
  if (g == 0) {
    float a2 = B1[d];
    for (int c = 0; c < 64; ++c) a2 += W1[d * 64 + c] * hv[c];
    h1[d] = 0.5f * a2 * (1.f + erff(a2 * 0.70710678118f));   // exact gelu
  }
  __syncthreads();
  if (g == 0) {
    float o = B2[d];
    for (int c = 0; c < 64; ++c) o += W2[d * 64 + c] * h1[c];
    slots[(size_t)(b * 8 + s) * 64 + d] = snew[d] + o;
  }
}

// ---------------------------------------------------------------------------
// Kernel D (last iter): slot colors (LN feat_color), position tanh-clip via
// aw @ W_pos^T, copy attn, assemble outputs: slots(8,8,72)|attn(8,8,4096)|pos.
// ---------------------------------------------------------------------------
__global__ __launch_bounds__(256) void k_final(
    const float* __restrict__ attn, const float* __restrict__ S,
    const float* __restrict__ Gx, const float* __restrict__ Gy,
    const float* __restrict__ feat_color,
    const float* __restrict__ g_color, const float* __restrict__ b_color,
    const float* __restrict__ W_pos, const float* __restrict__ b_pos,
    const float* __restrict__ slots, const float* __restrict__ pos,
    float* __restrict__ out)
{
  const int b = blockIdx.x >> 3;
  const int s = blockIdx.x & 7;
  const int tid = threadIdx.x;
  __shared__ float red[256];
  const float* at = attn + (size_t)(b * 8 + s) * NPIX;
  const float invS = 1.0f / S[b * 8 + s];
  float accC[8] = {0,0,0,0,0,0,0,0};
  float accP0 = 0.f, accP1 = 0.f;
  for (int n = tid; n < NPIX; n += 256) {
    float a = at[n] * invS;
    const float* fcr = feat_color + ((size_t)b * NPIX + n) * 8;
    float fc[8];
    float m = 0.f;
    for (int c = 0; c < 8; ++c) { fc[c] = fcr[c]; m += fc[c]; }
    m *= 0.125f;
    float v = 0.f;
    for (int c = 0; c < 8; ++c) { float t = fc[c] - m; v += t * t; }
    v *= 0.125f;
    float inv = 1.0f / sqrtf(v + 1e-5f);
    for (int c = 0; c < 8; ++c)
      accC[c] += a * ((fc[c] - m) * inv * g_color[c] + b_color[c]);
    if (s > 0) { accP0 += a * W_pos[n]; accP1 += a * W_pos[NPIX + n]; }
  }
  float C[8];
  for (int c = 0; c < 8; ++c) C[c] = block_reduce_256(accC[c], red, tid);
  float P0 = block_reduce_256(accP0, red, tid);
  float P1 = block_reduce_256(accP1, red, tid);
  if (tid < 64)
    out[(size_t)(b * 8 + s) * 72 + tid] = slots[(size_t)(b * 8 + s) * 64 + tid];
  else if (tid < 72)
    out[(size_t)(b * 8 + s) * 72 + tid] = C[tid - 64];
  float* oa = out + 4608 + (size_t)(b * 8 + s) * NPIX;
  for (int n = tid; n < NPIX; n += 256) oa[n] = at[n];
  if (s > 0 && tid == 0) {
    float mxp = Gx[b * 8 + s] * invS;
    float myp = Gy[b * 8 + s] * invS;
    float px = 0.5f * mxp + 0.5f * pos[(b * 7 + s - 1) * 2 + 0];
    float py = 0.5f * myp + 0.5f * pos[(b * 7 + s - 1) * 2 + 1];
    px += tanhf(P0 + b_pos[0]) * 0.1f;
    py += tanhf(P1 + b_pos[1]) * 0.1f;
    px = fminf(1.f, fmaxf(-1.f, px));
    py = fminf(1.f, fmaxf(-1.f, py));
    out[266752 + (b * 7 + s - 1) * 2 + 0] = px;
    out[266752 + (b * 7 + s - 1) * 2 + 1] = py;
  }
}

// ---------------------------------------------------------------------------
extern "C" void kernel_launch(void* const* d_in, const int* in_sizes, int n_in,
                              void* d_out, int out_size, void* d_ws, size_t ws_size,
                              hipStream_t stream) {
  (void)in_sizes; (void)n_in; (void)out_size; (void)ws_size;
  const float* feat       = (const float*)d_in[0];
  // d_in[1] camera_modulation: unused by the reference
  const float* feat_color = (const float*)d_in[2];
  const float* W_grid     = (const float*)d_in[3];
  const float* b_grid     = (const float*)d_in[4];
  const float* W_k_fg     = (const float*)d_in[5];
  const float* W_v_fg     = (const float*)d_in[6];
  const float* W_k_bg     = (const float*)d_in[7];
  const float* W_v_bg     = (const float*)d_in[8];
  const float* W_mlp_fg   = (const float*)d_in[9];
  const float* W_mlp_bg   = (const float*)d_in[10];
  const float* pinit      = (const float*)d_in[11];
  const float* sinit_fg   = (const float*)d_in[12];
  const float* sinit_bg   = (const float*)d_in[13];
  const float* W_q_fg     = (const float*)d_in[14];
  const float* W_q_bg     = (const float*)d_in[15];
  const float* g_q_fg     = (const float*)d_in[16];
  const float* b_q_fg     = (const float*)d_in[17];
  const float* g_q_bg     = (const float*)d_in[18];
  const float* b_q_bg     = (const float*)d_in[19];
  const float* g_feat     = (const float*)d_in[20];
  const float* b_feat     = (const float*)d_in[21];
  const float* g_color    = (const float*)d_in[22];
  const float* b_color    = (const float*)d_in[23];
  const float* gm_fg      = (const float*)d_in[24];
  const float* bm_fg      = (const float*)d_in[25];
  const float* gm_bg      = (const float*)d_in[26];
  const float* bm_bg      = (const float*)d_in[27];
  const float* W1f        = (const float*)d_in[28];
  const float* b1f        = (const float*)d_in[29];
  const float* W2f        = (const float*)d_in[30];
  const float* b2f        = (const float*)d_in[31];
  const float* W1b        = (const float*)d_in[32];
  const float* b1b        = (const float*)d_in[33];
  const float* W2b        = (const float*)d_in[34];
  const float* b2b        = (const float*)d_in[35];
  const float* W_pos      = (const float*)d_in[36];
  const float* b_pos      = (const float*)d_in[37];

  float* ws = (float*)d_ws;
  size_t o = 0;
  float* kf    = ws + o; o += (size_t)NB * NPIX * 64;   // 8 MB
  float* vfp   = ws + o; o += (size_t)NB * NPIX * 64;
  float* kbp   = ws + o; o += (size_t)NB * NPIX * 64;
  float* vbp   = ws + o; o += (size_t)NB * NPIX * 64;
  float* attn  = ws + o; o += (size_t)NB * 8 * NPIX;    // 1 MB
  float* qt    = ws + o; o += NB * 8 * 64;
  float* abc   = ws + o; o += 256;
  float* S     = ws + o; o += 64;
  float* Gxv   = ws + o; o += 64;
  float* Gyv   = ws + o; o += 64;
  float* slots = ws + o; o += NB * 8 * 64;
  float* pos   = ws + o; o += 128;
  float* Sp    = ws + o; o += NB * 16 * 8;
  float* Gxp   = ws + o; o += NB * 16 * 8;
  float* Gyp   = ws + o; o += NB * 16 * 8;

  k_pre<<<256, 256, 0, stream>>>(feat, g_feat, b_feat,
                                 W_k_fg, W_v_fg, W_k_bg, W_v_bg,
                                 kf, vfp, kbp, vbp);
  for (int it = 0; it < 4; ++it) {
    k_qt<<<64, 64, 0, stream>>>(W_q_fg, W_q_bg, W_mlp_fg, W_mlp_bg,
                                g_q_fg, b_q_fg, g_q_bg, b_q_bg,
                                W_grid, b_grid, sinit_fg, sinit_bg, pinit,
                                slots, pos, qt, abc, it);
    k_attn<<<128, 256, 0, stream>>>(kf, kbp, qt, abc, pos, attn, Sp, Gxp, Gyp);
    k_attn_red<<<1, 64, 0, stream>>>(Sp, Gxp, Gyp, S, Gxv, Gyv);
    if (it < 3)
      k_update<<<64, 256, 0, stream>>>(vfp, vbp, attn, S, Gxv, Gyv,
                                       W_grid, b_grid, W_mlp_fg, W_mlp_bg,
                                       gm_fg, bm_fg, gm_bg, bm_bg,
                                       W1f, b1f, W2f, b2f,
                                       W1b, b1b, W2b, b2b, slots, pos);
  }
  k_final<<<64, 256, 0, stream>>>(attn, S, Gxv, Gyv, feat_color,
                                  g_color, b_color, W_pos, b_pos,
                                  slots, pos, (float*)d_out);
}